// SpatialGradientIntegration_38199439130989
// MI455X (gfx1250) — compile-verified
//
#include <hip/hip_runtime.h>
#include <hip/hip_bf16.h>

typedef __attribute__((ext_vector_type(2))) float v2f;
typedef __attribute__((ext_vector_type(8))) float v8f;

#define IMG_W 1024
#define IMG_B 32

// -------------------------------------------------------------------------
// Channel 1: inclusive cumsum over HEIGHT via triangular WMMA.
//   For a 16-row segment:  Y = L * X + broadcast(carry),  L = lower-tri ones.
//   K=16 is split into 4 chained V_WMMA_F32_16X16X4_F32 ops.
//   One wave owns a 16-column strip over all 1024 rows (64 segments).
// -------------------------------------------------------------------------
__global__ __launch_bounds__(256) void cumsum_h_wmma(const float* __restrict__ in,
                                                     float* __restrict__ out) {
  const int lane = threadIdx.x & 31;
  const int wid  = threadIdx.x >> 5;
  const int g = blockIdx.x * 8 + wid;   // strip id 0..2047
  const int b = g >> 6;                 // image 0..31
  const int s = g & 63;                 // 16-col strip 0..63

  const size_t base = ((size_t)b * 2 + 1) * (size_t)IMG_W * IMG_W + (size_t)s * 16;
  const float* __restrict__ src = in + base;
  float* __restrict__ dst = out + base;

  const int n    = lane & 15;   // column within strip (B/C/D lane)
  const int half = lane >> 4;   // 0: lanes 0-15, 1: lanes 16-31
  const int m    = n;           // A-matrix row index (M = lane & 15)

  // Constant triangular A chunks, A_j = L[:, 4j:4j+4] in 16x4 A-layout:
  //   VGPR0 holds k' = 0 (lanes 0-15) / k' = 2 (lanes 16-31); VGPR1 holds k'=1/3.
  v2f a[4];
#pragma unroll
  for (int j = 0; j < 4; ++j) {
    const int k0 = 4 * j + 2 * half;
    a[j].x = (k0     <= m) ? 1.0f : 0.0f;
    a[j].y = (k0 + 1 <= m) ? 1.0f : 0.0f;
  }

  float carry = 0.0f;  // per-column running sum of all previous rows

  for (int seg = 0; seg < 64; ++seg) {
    const int R = seg * 16;

    // Prefetch the next segment (covers the serial carry dependency).
    if (seg < 63) {
      __builtin_prefetch(src + (size_t)(R + 16) * IMG_W + n, 0, 1);
      __builtin_prefetch(src + (size_t)(R + 24) * IMG_W + n, 0, 1);
    }

    // C = carry broadcast down the M dimension: every accumulator VGPR holds
    // carry[n] in lane n and lane n+16.
    v8f d;
#pragma unroll
    for (int i = 0; i < 8; ++i) d[i] = carry;

#pragma unroll
    for (int j = 0; j < 4; ++j) {
      // B_j = X[R+4j : R+4j+4, :] in 4x16 B-layout (mirrors A layout):
      //   VGPR0: row 4j (lanes 0-15) / row 4j+2 (lanes 16-31); VGPR1: rows 4j+1/4j+3.
      // Each component load = two contiguous 64B row chunks -> coalesced.
      v2f bj;
      bj.x = src[(size_t)(R + 4 * j + 2 * half    ) * IMG_W + n];
      bj.y = src[(size_t)(R + 4 * j + 2 * half + 1) * IMG_W + n];
      d = __builtin_amdgcn_wmma_f32_16x16x4_f32(
          /*neg_a=*/false, a[j], /*neg_b=*/false, bj,
          /*c_mod=*/(short)0, d, /*reuse_a=*/false, /*reuse_b=*/false);
    }

    // Store D: VGPR r = row R+r (lanes 0-15) / row R+r+8 (lanes 16-31).
#pragma unroll
    for (int r = 0; r < 8; ++r)
      dst[(size_t)(R + r + 8 * half) * IMG_W + n] = d[r];

    // New carry = row 15 of D = accumulator VGPR 7, lanes 16-31.
    carry = __shfl(d[7], 16 + n, 32);
  }
}

// -------------------------------------------------------------------------
// Channel 0: inclusive cumsum over WIDTH (contiguous axis).
//   One wave per row; 5-step shfl_up Hillis-Steele scan per 32-elem chunk,
//   lane-31 carry chained across 32 chunks. Fully coalesced 128B accesses.
// -------------------------------------------------------------------------
__global__ __launch_bounds__(256) void cumsum_w_scan(const float* __restrict__ in,
                                                     float* __restrict__ out) {
  const int lane = threadIdx.x & 31;
  const int wid  = threadIdx.x >> 5;
  const int r = blockIdx.x * 8 + wid;   // row id 0..32767
  const int b = r >> 10;                // image
  const int y = r & 1023;               // row within image

  const size_t base = ((size_t)b * 2) * (size_t)IMG_W * IMG_W + (size_t)y * IMG_W;
  const float* __restrict__ src = in + base;
  float* __restrict__ dst = out + base;

  float carry = 0.0f;
  for (int c = 0; c < 32; ++c) {
    float v = src[c * 32 + lane];
#pragma unroll
    for (int off = 1; off < 32; off <<= 1) {
      float t = __shfl_up(v, off, 32);
      if (lane >= off) v += t;
    }
    v += carry;
    dst[c * 32 + lane] = v;
    carry = __shfl(v, 31, 32);
  }
}

extern "C" void kernel_launch(void* const* d_in, const int* in_sizes, int n_in,
                              void* d_out, int out_size, void* d_ws, size_t ws_size,
                              hipStream_t stream) {
  (void)in_sizes; (void)n_in; (void)out_size; (void)d_ws; (void)ws_size;
  const float* in = (const float*)d_in[0];
  float* out = (float*)d_out;

  // Channel 0: 32768 rows, 8 waves (rows) per 256-thread block.
  cumsum_w_scan<<<4096, 256, 0, stream>>>(in, out);
  // Channel 1: 2048 column strips, 8 waves (strips) per 256-thread block.
  cumsum_h_wmma<<<256, 256, 0, stream>>>(in, out);
}